// Seiturd_32547262169353
// MI455X (gfx1250) — compile-verified
//
#include <hip/hip_runtime.h>
#include <math.h>

// Problem dims (fixed by the reference)
#define R_N 1024   // regions
#define D_N 128    // days
#define T_N 128    // steps

// d_out float offsets (tuple S,E,I,T,U,R,D concatenated flat)
#define OFF_S 0
#define OFF_E (T_N * R_N)
#define OFF_I (2 * T_N * R_N)
#define OFF_T (3 * T_N * R_N)
#define BIG   (T_N * D_N * R_N)
#define OFF_U (OFF_T + 1 * BIG)
#define OFF_R (OFF_T + 2 * BIG)
#define OFF_D (OFF_T + 3 * BIG)

typedef float v2f __attribute__((ext_vector_type(2)));
typedef float v8f __attribute__((ext_vector_type(8)));

// ---------------------------------------------------------------------------
// Barrier init (d_ws is not re-poisoned between replays -> reset every launch)
// ---------------------------------------------------------------------------
__global__ void init_bar(int* bar) { bar[0] = 0; bar[1] = 0; }

// ---------------------------------------------------------------------------
// Phase 1: sequential S/E/I recurrence with per-step 1024x1024 matvec.
// 32 persistent workgroups, each owns 32 rows of A; device-wide sync per step
// via an agent-scope atomic sense barrier. A (4MB) stays L2-resident.
// ---------------------------------------------------------------------------
#define NWG 32
#define ROWS_PER_WG (R_N / NWG)   // 32 rows, 8 waves -> 4 rows per wave

__global__ __launch_bounds__(256)
void phase1_seir(const float* __restrict__ A,
                 const float* __restrict__ lnE_p, const float* __restrict__ lnI_p,
                 const float* __restrict__ contagion, const int* __restrict__ tinit_p,
                 const float* __restrict__ S0, const float* __restrict__ E0,
                 const float* __restrict__ I0,
                 float* __restrict__ out, int* __restrict__ bar)
{
  __shared__ float Ivec[R_N];
  __shared__ float dotv[ROWS_PER_WG];

  const int tid = threadIdx.x;
  const int wg  = blockIdx.x;
  const int r0  = wg * ROWS_PER_WG;
  const float dE = expf(lnE_p[0]);
  const float dI = expf(lnI_p[0]);
  const float* cI = contagion + tinit_p[0] * R_N;   // day frozen at t_initial

  float S = 0.f, E = 0.f, I = 0.f;
  if (tid < ROWS_PER_WG) { S = S0[r0 + tid]; E = E0[r0 + tid]; I = I0[r0 + tid]; }

  const int wave = tid >> 5, lane = tid & 31;
  int gen = 0;

  for (int t = 0; t < T_N; ++t) {
    // Stage previous I vector (4KB) into LDS
    const float* src = (t == 0) ? I0 : (out + OFF_I + (size_t)(t - 1) * R_N);
    for (int i = tid; i < R_N; i += 256) Ivec[i] = src[i];
    __syncthreads();

    // Coalesced row-dot-products: one wave per row, float4 reads of A from L2
    #pragma unroll
    for (int i = 0; i < ROWS_PER_WG / 8; ++i) {
      const int row = r0 + wave * (ROWS_PER_WG / 8) + i;
      const float4* arow = (const float4*)(A + (size_t)row * R_N);
      float acc = 0.f;
      #pragma unroll
      for (int j = 0; j < 8; ++j) {
        float4 av = arow[lane + 32 * j];
        int k = 4 * (lane + 32 * j);
        acc += av.x * Ivec[k] + av.y * Ivec[k + 1] + av.z * Ivec[k + 2] + av.w * Ivec[k + 3];
      }
      #pragma unroll
      for (int m = 16; m; m >>= 1) acc += __shfl_xor(acc, m, 32);
      if (lane == 0) dotv[wave * (ROWS_PER_WG / 8) + i] = acc;
    }
    __syncthreads();

    if (tid < ROWS_PER_WG) {
      const int r = r0 + tid;
      float pSE = cI[r] * dotv[tid];
      float Sn = S - pSE * S;
      float En = E + pSE * S - dE * E;
      float In = I + dE * E - dI * I;
      S = Sn; E = En; I = In;
      out[OFF_S + (size_t)t * R_N + r] = S;
      out[OFF_E + (size_t)t * R_N + r] = E;
      out[OFF_I + (size_t)t * R_N + r] = I;
    }

    // Device-wide barrier (all 32 WGs co-resident: persistent launch)
    __threadfence();
    __syncthreads();
    if (tid == 0) {
      int n = __hip_atomic_fetch_add(&bar[0], 1, __ATOMIC_ACQ_REL, __HIP_MEMORY_SCOPE_AGENT);
      if (n == NWG - 1) {
        __hip_atomic_store(&bar[0], 0, __ATOMIC_RELAXED, __HIP_MEMORY_SCOPE_AGENT);
        __hip_atomic_fetch_add(&bar[1], 1, __ATOMIC_RELEASE, __HIP_MEMORY_SCOPE_AGENT);
      } else {
        while (__hip_atomic_load(&bar[1], __ATOMIC_ACQUIRE, __HIP_MEMORY_SCOPE_AGENT) <= gen) {
          __builtin_amdgcn_s_sleep(1);
        }
      }
      ++gen;
    }
    __syncthreads();
  }
}

// ---------------------------------------------------------------------------
// Phase 2: T/U/R/D trajectories via scan-as-matmul.
//   G = W  @ ITRAJ   (W[t,s] = a^{t-s}, s<=t)         a = 1 - decay_T
//   C = L  @ ITRAJ   (ones lower-tri)
//   H = LW @ ITRAJ   (LW[t,s] = geo(t-1-s), s<t)
// Weight tiles are materialized in LDS once so the K-loop is pure
// ds_load + v_wmma_f32_16x16x4_f32 with uniform control flow, then the
// streaming phase does the 270MB of coalesced float4 stores (roofline term).
// ---------------------------------------------------------------------------
#define TTL 16
#define RRL 64
#define ISTR 65           // LDS pad to dodge bank conflicts (scalar reads)
#define TSTR 68           // LDS pad keeping 16B alignment (float4 reads)
#define P2_THREADS 384    // 12 wave32s: 3 matrices x 4 N-tiles

__global__ __launch_bounds__(P2_THREADS)
void phase2_turd(const float* __restrict__ lnI_p, const float* __restrict__ lnT_p,
                 const float* __restrict__ det, const float* __restrict__ rec,
                 const float* __restrict__ T0, const float* __restrict__ U0,
                 const float* __restrict__ R0, const float* __restrict__ D0,
                 float* __restrict__ out)
{
  __shared__ __align__(16) float Itile[T_N * ISTR];       // ITRAJ[:, 64 cols]
  __shared__ __align__(16) float Wt[3 * T_N * TTL];       // [mat][s][m] weights
  __shared__ __align__(16) float Gt[TTL * TSTR];
  __shared__ __align__(16) float Ct[TTL * TSTR];
  __shared__ __align__(16) float Ht[TTL * TSTR];
  __shared__ float apow[T_N + 1];
  __shared__ float geos[T_N];
  __shared__ float T0s[RRL], U0s[RRL], R0s[RRL], D0s[RRL];

  const int tid = threadIdx.x;
  const int t0  = blockIdx.x * TTL;
  const int rb  = blockIdx.y * RRL;
  const float dI = expf(lnI_p[0]);
  const float dT = expf(lnT_p[0]);

  // Stage ITRAJ column tile (128 x 64) from d_out into LDS
  const float* itraj = out + OFF_I;
  for (int idx = tid; idx < T_N * RRL; idx += P2_THREADS) {
    int s = idx >> 6, c = idx & 63;
    Itile[s * ISTR + c] = itraj[(size_t)s * R_N + rb + c];
  }
  for (int c = tid; c < RRL; c += P2_THREADS) {
    T0s[c] = T0[rb + c]; U0s[c] = U0[rb + c];
    R0s[c] = R0[rb + c]; D0s[c] = D0[rb + c];
  }
  if (tid == 0) {  // powers + geometric sums in double (cheap, per block)
    double a = 1.0 - (double)dT, p = 1.0, g = 0.0;
    for (int s = 0; s < T_N; ++s) { apow[s] = (float)p; g += p; geos[s] = (float)g; p *= a; }
    apow[T_N] = (float)p;
  }
  __syncthreads();

  // Materialize the three 16x128 weight tiles once: Wt[mat][s][m], m = t-local.
  for (int idx = tid; idx < 3 * T_N * TTL; idx += P2_THREADS) {
    const int m   = idx & 15;
    const int s   = (idx >> 4) & (T_N - 1);
    const int mat = idx >> 11;
    const int ta  = t0 + m;
    float w;
    if (mat == 0)      w = (s <= ta) ? apow[ta - s] : 0.f;
    else if (mat == 1) w = (s <= ta) ? 1.f : 0.f;
    else               w = (s <  ta) ? geos[ta - 1 - s] : 0.f;
    Wt[idx] = w;
  }
  __syncthreads();

  const int wave = tid >> 5, lane = tid & 31;
  {
    // One (matrix, N-tile) job per wave; readfirstlane => scalar selection.
    const int mat = __builtin_amdgcn_readfirstlane(wave >> 2);   // 0:G 1:C 2:H
    const int nt  = __builtin_amdgcn_readfirstlane(wave & 3);    // N-tile
    const int m   = lane & 15;       // A: M row / B,C: N column
    const int kh  = lane >> 4;       // half-wave K offset
    const float* wbase = Wt + mat * (T_N * TTL);
    v8f acc = {0.f, 0.f, 0.f, 0.f, 0.f, 0.f, 0.f, 0.f};
    #pragma unroll 4
    for (int kk = 0; kk < T_N / 4; ++kk) {
      const int sA = 4 * kk + 2 * kh;          // K = 2*kh + j  (A/B layouts)
      v2f Aop;
      Aop[0] = wbase[(sA    ) * TTL + m];
      Aop[1] = wbase[(sA + 1) * TTL + m];
      v2f Bop;
      Bop[0] = Itile[(sA    ) * ISTR + nt * 16 + m];
      Bop[1] = Itile[(sA + 1) * ISTR + nt * 16 + m];
      acc = __builtin_amdgcn_wmma_f32_16x16x4_f32(false, Aop, false, Bop,
                                                  (short)0, acc, false, false);
    }
    float* dst = (mat == 0) ? Gt : (mat == 1) ? Ct : Ht;
    #pragma unroll
    for (int j = 0; j < 8; ++j)   // C layout: VGPR j -> M = j + 8*kh, N = m
      dst[(j + 8 * kh) * TSTR + nt * 16 + m] = acc[j];
  }
  __syncthreads();

  // Streaming phase: all 128 days for this (16t x 64r) tile. Store-bound.
  const int QUADS = TTL * D_N * (RRL / 4);   // 32768 float4 groups
  for (int iq = tid; iq < QUADS; iq += P2_THREADS) {
    const int q  = iq & 15;
    const int dd = (iq >> 4) & 127;
    const int tt = iq >> 11;
    const int t_abs = t0 + tt;
    const int rr = q * 4;
    const size_t colg = (size_t)rb + rr;

    // Prefetch next iteration's rate tiles (global_prefetch_b8 on gfx1250)
    const int iqn = iq + P2_THREADS;
    if (iqn < QUADS) {
      const size_t pref = (size_t)((iqn >> 4) & 127) * R_N + rb + (iqn & 15) * 4;
      __builtin_prefetch(det + pref, 0, 1);
      __builtin_prefetch(rec + pref, 0, 1);
    }

    const float4 det4 = *(const float4*)(det + (size_t)dd * R_N + colg);
    const float4 rec4 = *(const float4*)(rec + (size_t)dd * R_N + colg);
    const float4 g4 = *(const float4*)(Gt + tt * TSTR + rr);
    const float4 c4 = *(const float4*)(Ct + tt * TSTR + rr);
    const float4 h4 = *(const float4*)(Ht + tt * TSTR + rr);
    const float aT1 = apow[t_abs + 1];
    const float geo = geos[t_abs];

    float4 To, Uo, Ro, Do;
    auto comp = [&](float dv, float rv, float gv, float cv, float hv, int rl,
                    float& Tv, float& Uv, float& Rv, float& Dv) {
      const float pIT = dv * dI;
      const float pIU = (1.f - dv) * dI;
      const float pTR = rv * dT;
      const float pTD = (1.f - rv) * dT;
      const float t0v = T0s[rl];
      Tv = aT1 * t0v + pIT * gv;
      Uv = U0s[rl] + pIU * cv;
      const float scum = t0v * geo + pIT * hv;
      Rv = R0s[rl] + pTR * scum;
      Dv = D0s[rl] + pTD * scum;
    };
    comp(det4.x, rec4.x, g4.x, c4.x, h4.x, rr + 0, To.x, Uo.x, Ro.x, Do.x);
    comp(det4.y, rec4.y, g4.y, c4.y, h4.y, rr + 1, To.y, Uo.y, Ro.y, Do.y);
    comp(det4.z, rec4.z, g4.z, c4.z, h4.z, rr + 2, To.z, Uo.z, Ro.z, Do.z);
    comp(det4.w, rec4.w, g4.w, c4.w, h4.w, rr + 3, To.w, Uo.w, Ro.w, Do.w);

    const size_t base = ((size_t)t_abs * D_N + dd) * R_N + colg;
    *(float4*)(out + OFF_T + base) = To;
    *(float4*)(out + OFF_U + base) = Uo;
    *(float4*)(out + OFF_R + base) = Ro;
    *(float4*)(out + OFF_D + base) = Do;
  }
}

// ---------------------------------------------------------------------------
extern "C" void kernel_launch(void* const* d_in, const int* in_sizes, int n_in,
                              void* d_out, int out_size, void* d_ws, size_t ws_size,
                              hipStream_t stream) {
  const float* A    = (const float*)d_in[0];
  const float* lnE  = (const float*)d_in[1];
  const float* lnI  = (const float*)d_in[2];
  const float* lnT  = (const float*)d_in[3];
  const float* det  = (const float*)d_in[4];
  const float* rec  = (const float*)d_in[5];
  const float* cont = (const float*)d_in[6];
  const float* S0   = (const float*)d_in[7];
  const float* E0   = (const float*)d_in[8];
  const float* I0   = (const float*)d_in[9];
  const float* T0   = (const float*)d_in[10];
  const float* U0   = (const float*)d_in[11];
  const float* R0   = (const float*)d_in[12];
  const float* D0   = (const float*)d_in[13];
  const int*   tin  = (const int*)d_in[14];
  float* out = (float*)d_out;
  int* bar = (int*)d_ws;

  init_bar<<<1, 1, 0, stream>>>(bar);
  phase1_seir<<<NWG, 256, 0, stream>>>(A, lnE, lnI, cont, tin, S0, E0, I0, out, bar);
  phase2_turd<<<dim3(T_N / TTL, R_N / RRL), P2_THREADS, 0, stream>>>(
      lnI, lnT, det, rec, T0, U0, R0, D0, out);
}